// HashCodeAwareLogits_61761629716552
// MI455X (gfx1250) — compile-verified
//
#include <hip/hip_runtime.h>
#include <hip/hip_bf16.h>

// ---------------------------------------------------------------------------
// HashCodeAwareLogits for MI455X (gfx1250).
// Memory-bound: ~1.07 GB of random 8KB bucket-row gathers -> ~46us HBM floor.
// Strategy:
//   kernel 1: per-b prefix hashes -> importance weights + bucket ids (workspace)
//   kernel 2: per (b,d,h): TDM tensor_load_to_lds of the 8KB bucket row
//             (double buffered, TENSORcnt), TDM pad-insertion (stride 67 dw)
//             for bank-conflict-free ds_load_b64 in WMMA A layout, then
//             v_wmma_f32_16x16x4_f32 chains compute the 32x64 matvec.
// ---------------------------------------------------------------------------

typedef float v2f __attribute__((ext_vector_type(2)));
typedef float v8f __attribute__((ext_vector_type(8)));
typedef unsigned int v4u __attribute__((ext_vector_type(4)));
typedef int v4i __attribute__((ext_vector_type(4)));
typedef int v8i __attribute__((ext_vector_type(8)));

#define PRIME_LL 2147483647LL
#define N_DIGITS 16
#define N_ARY 32
#define EMB 64
#define NUM_EMB 100000
#define ROW_ELEMS 2048            // N_ARY * EMB floats per bucket row
#define STRIDE_DW 67              // 64 data dwords + 3 TDM pad dwords per a-row
#define ROW_DW (N_ARY * STRIDE_DW) // 2144 dwords per padded row in LDS

struct HashConsts {
    long long seq_a[N_DIGITS];
    long long ha[2];
    long long hb[2];
};

// ---------------------------------------------------------------------------
// Stage 1: prefix hashes -> (w0,w1) and (bucket0,bucket1) per (b,d)
// ---------------------------------------------------------------------------
__global__ void hash_stage(const long long* __restrict__ seq,
                           const float* __restrict__ iw,
                           float2* __restrict__ w_out,
                           uint2* __restrict__ bkt_out,
                           HashConsts hc, int B) {
    int b = blockIdx.x * blockDim.x + threadIdx.x;
    if (b >= B) return;
    long long h[N_DIGITS];
    long long acc = 0;
    int len = 0;
    for (int i = 0; i < N_DIGITS; ++i) {
        long long s = seq[(size_t)b * N_DIGITS + i];
        if (s != 0) len = i + 1;
        acc = (acc + hc.seq_a[i] * (s % PRIME_LL)) % PRIME_LL;
        h[i] = acc;
    }
    int last = (len > 0) ? (len - 1) : 0;
    for (int d = 0; d < N_DIGITS; ++d) {
        int idx = d < last ? d : last;
        long long id = h[idx];
        int widx = (int)(id % (long long)NUM_EMB);
        float2 w;
        w.x = iw[widx * 2 + 0];
        w.y = iw[widx * 2 + 1];
        w_out[(size_t)b * N_DIGITS + d] = w;
        uint2 bk;
        bk.x = (unsigned)(((hc.ha[0] * id + hc.hb[0]) % PRIME_LL) & 65535LL);
        bk.y = (unsigned)(((hc.ha[1] * id + hc.hb[1]) % PRIME_LL) & 65535LL);
        bkt_out[(size_t)b * N_DIGITS + d] = bk;
    }
}

// ---------------------------------------------------------------------------
// TDM: one-instruction DMA of a 2048-f32 bucket row into LDS with pad
// insertion (3 dwords after every 64) so LDS a-row stride is 67 dwords.
// ---------------------------------------------------------------------------
__device__ __forceinline__ void tdm_row_load(const float* gptr, unsigned lds_byte) {
    unsigned long long ga = (unsigned long long)gptr;
    unsigned galo = (unsigned)__builtin_amdgcn_readfirstlane((int)(ga & 0xffffffffULL));
    unsigned gahi = (unsigned)__builtin_amdgcn_readfirstlane((int)(ga >> 32));
    unsigned lb   = (unsigned)__builtin_amdgcn_readfirstlane((int)lds_byte);

    // D# group0: count=1, lds_addr, global_addr[56:0], type=2 ("image")
    v4u g0;
    g0.x = 1u;
    g0.y = lb;
    g0.z = galo;
    g0.w = (gahi & 0x1ffffffu) | (2u << 30);

    // D# group1: data_size=4B, pad_enable=1, pad_interval=64dw, pad_amount=3dw,
    //            tensor_dim0=2048, tensor_dim1=1, tile_dim0=2048, tile_dim1=1,
    //            tensor_dim0_stride=2048
    v8i g1;
    g1[0] = (int)((2u << 16) | (1u << 20) | (5u << 22) | (2u << 25));
    g1[1] = (int)(0x0800u << 16);   // tensor_dim0[15:0]=2048 in bits[31:16]
    g1[2] = (int)(1u << 16);        // tensor_dim0 hi = 0, tensor_dim1 = 1
    g1[3] = (int)(0x0800u << 16);   // tile_dim0 = 2048
    g1[4] = 1;                      // tile_dim1 = 1, tile_dim2 = 0
    g1[5] = 2048;                   // tensor_dim0_stride
    g1[6] = 0;
    g1[7] = 0;

    v4i z4 = {0, 0, 0, 0};
#if defined(__clang_major__) && (__clang_major__ >= 23)
    v8i z8 = {0, 0, 0, 0, 0, 0, 0, 0};
    __builtin_amdgcn_tensor_load_to_lds(g0, g1, z4, z4, z8, 0);
#else
    __builtin_amdgcn_tensor_load_to_lds(g0, g1, z4, z4, 0);
#endif
}

// ---------------------------------------------------------------------------
// Stage 2: one block per b, 2 waves, 8 digits per wave, TDM double-buffered.
// out[b,d,a] = sum_h w_h * sum_e table[bucket_h][a*64+e] * rep[b,d,e]
// via v_wmma_f32_16x16x4_f32 (B = rep chunk replicated across N columns).
// ---------------------------------------------------------------------------
__global__ void __launch_bounds__(64) gather_dot(
        const float* __restrict__ table,
        const float* __restrict__ rep,
        const float2* __restrict__ wv,
        const uint2* __restrict__ bkt,
        float* __restrict__ out) {
    const int b = blockIdx.x;
    const int wave = threadIdx.x >> 5;
    const int lane = threadIdx.x & 31;
    const int mlane = lane & 15;
    const int hi = lane >> 4;          // which half of the wave
    const int hi2 = hi * 2;            // K sub-offset for A/B layout

    __shared__ float lds[2][2][ROW_DW];   // [wave][buffer][padded row]

    auto issue = [&](int r) {
        int dd = r >> 1;
        int h = r & 1;
        int d = wave * 8 + dd;
        uint2 bb = bkt[(size_t)b * N_DIGITS + d];
        unsigned bucket = h ? bb.y : bb.x;
        const float* gp = table + (size_t)bucket * ROW_ELEMS;
        unsigned lbyte = (unsigned)(unsigned long long)(&lds[wave][r & 1][0]);
        tdm_row_load(gp, lbyte);
    };

    issue(0);
    issue(1);

    for (int dd = 0; dd < 8; ++dd) {
        const int d = wave * 8 + dd;
        const float* repp = rep + ((size_t)b * N_DIGITS + d) * EMB;

        // Pre-stage the B operand: rep K-chunks in WMMA B layout
        // (lane half hi supplies rows K = 2*hi, 2*hi+1 of each 4x16 tile).
        v2f bvs[16];
#pragma unroll
        for (int kc = 0; kc < 16; ++kc)
            bvs[kc] = *(const v2f*)(repp + kc * 4 + hi2);

        const float2 wpair = wv[(size_t)b * N_DIGITS + d];
        v8f acc0 = {};
        v8f acc1 = {};

        for (int h = 0; h < 2; ++h) {
            const int r = dd * 2 + h;
            if (r >= 14) {
                __builtin_amdgcn_s_wait_tensorcnt(0);
            } else {
                __builtin_amdgcn_s_wait_tensorcnt(1);
            }
            const float* buf = &lds[wave][r & 1][0];

            v8f c0 = {};
            v8f c1 = {};
#pragma unroll
            for (int kc = 0; kc < 16; ++kc) {
                // A layout: lane m holds A[m, kc*4 + hi*2 .. +1]; stride 67
                // dwords between a-rows -> conflict-free ds_load_b64.
                v2f a0 = *(const v2f*)(buf + mlane * STRIDE_DW + kc * 4 + hi2);
                v2f a1 = *(const v2f*)(buf + (mlane + 16) * STRIDE_DW + kc * 4 + hi2);
                c0 = __builtin_amdgcn_wmma_f32_16x16x4_f32(
                        false, a0, false, bvs[kc], (short)0, c0, false, false);
                c1 = __builtin_amdgcn_wmma_f32_16x16x4_f32(
                        false, a1, false, bvs[kc], (short)0, c1, false, false);
            }
            const float wgt = h ? wpair.y : wpair.x;
            acc0 += wgt * c0;
            acc1 += wgt * c1;

            if (r + 2 < 16) issue(r + 2);
        }

        // C layout: VGPR v of lane-half hi holds D[M = 8*hi + v, n] (all n
        // columns equal). Lanes 0 and 16 write 8 contiguous a's each.
        if (mlane == 0) {
            float* o = out + ((size_t)b * N_DIGITS + d) * N_ARY + hi * 8;
            *(v8f*)(o) = acc0;        // a-block 0
            *(v8f*)(o + 16) = acc1;   // a-block 1
        }
    }
}

// ---------------------------------------------------------------------------
// Host-side replication of numpy RandomState(42) draws for SEQ_A/HASH_A/HASH_B
// (MT19937 + legacy masked-rejection bounded sampling, next64 = hi<<32|lo).
// ---------------------------------------------------------------------------
namespace {
struct MT19937 {
    unsigned int mt[624];
    int mti;
    void seed(unsigned int s) {
        mt[0] = s;
        for (int i = 1; i < 624; ++i)
            mt[i] = 1812433253u * (mt[i - 1] ^ (mt[i - 1] >> 30)) + (unsigned)i;
        mti = 624;
    }
    unsigned int next32() {
        if (mti >= 624) {
            for (int i = 0; i < 624; ++i) {
                unsigned int y = (mt[i] & 0x80000000u) | (mt[(i + 1) % 624] & 0x7fffffffu);
                mt[i] = mt[(i + 397) % 624] ^ (y >> 1) ^ ((y & 1u) ? 2567483615u : 0u);
            }
            mti = 0;
        }
        unsigned int y = mt[mti++];
        y ^= y >> 11;
        y ^= (y << 7) & 2636928640u;
        y ^= (y << 15) & 4022730752u;
        y ^= y >> 18;
        return y;
    }
    unsigned long long next64() {
        unsigned long long hiw = next32();
        unsigned long long low = next32();
        return (hiw << 32) | low;
    }
    long long randint(long long lo, long long hiex) {  // numpy randint [lo, hiex)
        unsigned long long rng = (unsigned long long)(hiex - 1 - lo);
        unsigned long long mask = rng;
        mask |= mask >> 1;  mask |= mask >> 2;  mask |= mask >> 4;
        mask |= mask >> 8;  mask |= mask >> 16; mask |= mask >> 32;
        unsigned long long v;
        do { v = next64() & mask; } while (v > rng);
        return lo + (long long)v;
    }
};
} // namespace

extern "C" void kernel_launch(void* const* d_in, const int* in_sizes, int n_in,
                              void* d_out, int out_size, void* d_ws, size_t ws_size,
                              hipStream_t stream) {
    const long long* seq  = (const long long*)d_in[0];  // int64 sense codes
    const float* rep      = (const float*)d_in[1];      // [B,16,64] f32
    const float* iw       = (const float*)d_in[2];      // [100000,2] f32
    const float* table    = (const float*)d_in[3];      // [65536,2048] f32
    float* out            = (float*)d_out;              // [B,16,32] f32

    const int B = in_sizes[0] / N_DIGITS;

    HashConsts hc;
    MT19937 mt;
    mt.seed(42u);
    for (int i = 0; i < N_DIGITS; ++i) hc.seq_a[i] = mt.randint(1, PRIME_LL);
    for (int i = 0; i < 2; ++i)        hc.ha[i]    = mt.randint(1, PRIME_LL);
    for (int i = 0; i < 2; ++i)        hc.hb[i]    = mt.randint(0, PRIME_LL);

    float2* w_ws  = (float2*)d_ws;
    uint2* bkt_ws = (uint2*)((char*)d_ws + (size_t)B * N_DIGITS * sizeof(float2));

    hipLaunchKernelGGL(hash_stage, dim3((B + 255) / 256), dim3(256), 0, stream,
                       seq, iw, w_ws, bkt_ws, hc, B);
    hipLaunchKernelGGL(gather_dot, dim3(B), dim3(64), 0, stream,
                       table, rep, w_ws, bkt_ws, out);
}